// GCN_Adj_8581344658003
// MI455X (gfx1250) — compile-verified
//
#include <hip/hip_runtime.h>
#include <hip/hip_bf16.h>

typedef float v2f __attribute__((ext_vector_type(2)));
typedef float v8f __attribute__((ext_vector_type(8)));

#define N_NODES   50000
#define N_EDGES   800000
#define IN_FEATS  512
#define N_HIDDEN  128
#define N_CLASSES 40
#define KCHUNK    32
#define A_PAD     34   // padded LDS row stride (floats), even -> 8B-aligned b64 loads

// ---------------------------------------------------------------------------
// GEMM1: H1[m][n] = (sum_k F[m][k] * W1[k][n]) * norm[m]
// Block = 256 threads = 8 waves; each wave: one 16-row tile, all 128 cols.
// B panel stored k-pair interleaved: fragment {B[k][c],B[k+1][c]} is one b64.
// ---------------------------------------------------------------------------
__global__ __launch_bounds__(256) void gemm1_wmma(
    const float* __restrict__ F, const float* __restrict__ W1,
    const float* __restrict__ norm, float* __restrict__ H1) {
  __shared__ float sB[KCHUNK * N_HIDDEN];   // [k>>1][col][k&1]   16 KB
  __shared__ float sA[128 * A_PAD];         // [row][k] padded    17 KB

  const int lane = threadIdx.x & 31;
  const int wave = threadIdx.x >> 5;        // 0..7
  const int half = lane >> 4;               // 0: K{0,1} / 1: K{2,3}
  const int l16  = lane & 15;
  const int blockRow0 = blockIdx.x * 128;
  const int m0 = blockRow0 + wave * 16;

  v8f acc[8] = {};

  for (int kc = 0; kc < IN_FEATS; kc += KCHUNK) {
    // stage B chunk (32x128) k-pair interleaved; global read stays coalesced
    for (int i = threadIdx.x; i < KCHUNK * N_HIDDEN; i += 256) {
      int k = i >> 7, col = i & 127;
      sB[(k >> 1) * (2 * N_HIDDEN) + col * 2 + (k & 1)] =
          W1[(size_t)(kc + k) * N_HIDDEN + col];
    }
    // stage A chunk (128 rows x 32 k), coalesced, padded stride
    for (int i = threadIdx.x; i < 128 * KCHUNK; i += 256) {
      int r = i >> 5, k = i & 31;
      int gr = blockRow0 + r;
      sA[r * A_PAD + k] = (gr < N_NODES) ? F[(size_t)gr * IN_FEATS + kc + k] : 0.f;
    }
    if (kc + KCHUNK < IN_FEATS)   // global_prefetch_b8 for next weight chunk
      __builtin_prefetch(&W1[(size_t)(kc + KCHUNK) * N_HIDDEN + threadIdx.x], 0, 0);
    __syncthreads();

    const int arow = wave * 16 + l16;
    #pragma unroll
    for (int ks = 0; ks < KCHUNK; ks += 4) {
      const int ak = ks + half * 2;           // even
      const int kp = ak >> 1;                 // k-pair row in sB
      v2f a = *(const v2f*)&sA[arow * A_PAD + ak];              // ds_load_b64
      #pragma unroll
      for (int t = 0; t < 8; t++) {
        v2f b = *(const v2f*)&sB[kp * (2 * N_HIDDEN) + (t * 16 + l16) * 2]; // b64
        acc[t] = __builtin_amdgcn_wmma_f32_16x16x4_f32(
            false, a, false, b, (short)0, acc[t], false, false);
      }
    }
    __syncthreads();
  }

  #pragma unroll
  for (int r = 0; r < 8; r++) {
    int m = m0 + r + half * 8;
    if (m < N_NODES) {
      float nm = norm[m];
      #pragma unroll
      for (int t = 0; t < 8; t++)
        H1[(size_t)m * N_HIDDEN + t * 16 + l16] = acc[t][r] * nm;
    }
  }
}

// ---------------------------------------------------------------------------
// GEMM2: H3[m][c] = (sum_k relu(H2[m][k]*norm[m]) * W2[k][c]) * norm[m]
// K = 128, N = 40 (padded to 48 -> 3 WMMA N-tiles). Same interleaved B panel.
// ---------------------------------------------------------------------------
__global__ __launch_bounds__(256) void gemm2_wmma(
    const float* __restrict__ H2, const float* __restrict__ W2,
    const float* __restrict__ norm, float* __restrict__ H3) {
  __shared__ float sB[N_HIDDEN * 48];   // [k>>1][col(48)][k&1]  24 KB

  const int lane = threadIdx.x & 31;
  const int wave = threadIdx.x >> 5;
  const int half = lane >> 4;
  const int l16  = lane & 15;

  for (int i = threadIdx.x; i < N_HIDDEN * 48; i += 256) {
    int k = i / 48, c = i % 48;
    sB[(k >> 1) * 96 + c * 2 + (k & 1)] =
        (c < N_CLASSES) ? W2[(size_t)k * N_CLASSES + c] : 0.f;
  }
  __syncthreads();

  const int m0 = (blockIdx.x * 8 + wave) * 16;
  v8f acc[3] = {};

  const int  arow  = m0 + l16;
  const bool rowok = arow < N_NODES;
  const float anorm = rowok ? norm[arow] : 0.f;   // 0 -> zero A-fragment for pad rows
  const float* Arow = H2 + (size_t)(rowok ? arow : 0) * N_HIDDEN;

  #pragma unroll 4
  for (int ks = 0; ks < N_HIDDEN; ks += 4) {
    const int ak = ks + half * 2;
    const int kp = ak >> 1;
    v2f raw = *(const v2f*)&Arow[ak];              // global_load_b64
    v2f a;
    a.x = fmaxf(raw.x * anorm, 0.f);
    a.y = fmaxf(raw.y * anorm, 0.f);
    #pragma unroll
    for (int t = 0; t < 3; t++) {
      v2f b = *(const v2f*)&sB[kp * 96 + (t * 16 + l16) * 2];   // ds_load_b64
      acc[t] = __builtin_amdgcn_wmma_f32_16x16x4_f32(
          false, a, false, b, (short)0, acc[t], false, false);
    }
  }

  #pragma unroll
  for (int r = 0; r < 8; r++) {
    int m = m0 + r + half * 8;
    if (m < N_NODES) {
      float nm = norm[m];
      #pragma unroll
      for (int t = 0; t < 3; t++) {
        int c = t * 16 + l16;
        if (c < N_CLASSES) H3[(size_t)m * N_CLASSES + c] = acc[t][r] * nm;
      }
    }
  }
}

// ---------------------------------------------------------------------------
// SpMM copy-sum via fp32 global atomics (working set fits in 192 MB L2)
// ---------------------------------------------------------------------------
__global__ __launch_bounds__(256) void spmm1_scatter(
    const float* __restrict__ H1, const int* __restrict__ src,
    const int* __restrict__ dst, float* __restrict__ H2) {
  int idx = blockIdx.x * 256 + threadIdx.x;     // (edge, feat)
  int e = idx >> 7, f = idx & 127;
  if (e < N_EDGES) {
    float v = H1[(size_t)src[e] * N_HIDDEN + f];
    unsafeAtomicAdd(&H2[(size_t)dst[e] * N_HIDDEN + f], v);
  }
}

__global__ __launch_bounds__(256) void spmm2_scatter(
    const float* __restrict__ H3, const int* __restrict__ src,
    const int* __restrict__ dst, float* __restrict__ OUT) {
  int idx = blockIdx.x * 256 + threadIdx.x;
  if (idx < N_EDGES * N_CLASSES) {
    int e = idx / N_CLASSES, f = idx % N_CLASSES;
    float v = H3[(size_t)src[e] * N_CLASSES + f];
    unsafeAtomicAdd(&OUT[(size_t)dst[e] * N_CLASSES + f], v);
  }
}

__global__ __launch_bounds__(256) void zero_f32(float* __restrict__ p, int n) {
  int i = blockIdx.x * 256 + threadIdx.x;
  if (i < n) p[i] = 0.f;
}

__global__ __launch_bounds__(256) void scale_rows(
    float* __restrict__ out, const float* __restrict__ norm) {
  int i = blockIdx.x * 256 + threadIdx.x;
  if (i < N_NODES * N_CLASSES) out[i] *= norm[i / N_CLASSES];
}

// ---------------------------------------------------------------------------
extern "C" void kernel_launch(void* const* d_in, const int* in_sizes, int n_in,
                              void* d_out, int out_size, void* d_ws, size_t ws_size,
                              hipStream_t stream) {
  (void)in_sizes; (void)n_in; (void)out_size; (void)ws_size;
  const float* F    = (const float*)d_in[0];
  const float* norm = (const float*)d_in[1];
  const float* W1   = (const float*)d_in[2];
  const float* W2   = (const float*)d_in[3];
  const int*   esrc = (const int*)d_in[4];
  const int*   edst = (const int*)d_in[5];
  float* out = (float*)d_out;

  float* ws = (float*)d_ws;
  float* H1 = ws;                                   // 50000*128
  float* H2 = ws + (size_t)N_NODES * N_HIDDEN;      // 50000*128
  float* H3 = ws;                                   // reuse H1 region (50000*40)

  const int gemmBlocks = (N_NODES + 127) / 128;     // 391

  zero_f32<<<(N_NODES * N_HIDDEN + 255) / 256, 256, 0, stream>>>(H2, N_NODES * N_HIDDEN);
  zero_f32<<<(N_NODES * N_CLASSES + 255) / 256, 256, 0, stream>>>(out, N_NODES * N_CLASSES);

  gemm1_wmma<<<gemmBlocks, 256, 0, stream>>>(F, W1, norm, H1);
  spmm1_scatter<<<(N_EDGES * N_HIDDEN + 255) / 256, 256, 0, stream>>>(H1, esrc, edst, H2);
  gemm2_wmma<<<gemmBlocks, 256, 0, stream>>>(H2, W2, norm, H3);
  spmm2_scatter<<<(N_EDGES * N_CLASSES + 255) / 256, 256, 0, stream>>>(H3, esrc, edst, out);
  scale_rows<<<(N_NODES * N_CLASSES + 255) / 256, 256, 0, stream>>>(out, norm);
}